// Focal_L2_83459804495948
// MI455X (gfx1250) — compile-verified
//
#include <hip/hip_runtime.h>

// ---------------------------------------------------------------------------
// Focal L2 loss, shape (32, 3, 512, 512) fp32.
// Phase 1: 96 channels x 32 blocks stream-reduce (sum_in, sum_tgt, sum_sqdiff)
//          with NT b128 loads -> unique workspace slots (deterministic).
// Phase 2: one wave32 block folds 96x32 partials into the scalar loss; the
//          final 32-lane reduction uses V_WMMA_F32_16X16X4_F32 with B = ones.
// ---------------------------------------------------------------------------

typedef float f32x4 __attribute__((ext_vector_type(4)));
typedef float v2f   __attribute__((ext_vector_type(2)));
typedef float v8f   __attribute__((ext_vector_type(8)));

#define HW            262144      // 512*512
#define NCH           96          // 32*3 channels
#define BLK_PER_CH    32
#define THREADS       256
#define VEC_PER_CH    (HW / 4)                 // 65536 float4 per channel
#define VEC_PER_BLK   (VEC_PER_CH / BLK_PER_CH) // 2048 float4 per block
#define ITERS         (VEC_PER_BLK / THREADS)   // 8
#define SLOTS         (NCH * BLK_PER_CH)        // 3072 partials per quantity

__global__ __launch_bounds__(THREADS)
void focal_partial(const float* __restrict__ inp,
                   const float* __restrict__ tgt,
                   float* __restrict__ ws) {
    const int ch = blockIdx.y;          // 0..95
    const int bx = blockIdx.x;          // 0..31
    const long base = (long)ch * VEC_PER_CH + (long)bx * VEC_PER_BLK + threadIdx.x;
    const f32x4* in4 = (const f32x4*)inp + base;
    const f32x4* tg4 = (const f32x4*)tgt + base;

    float si = 0.f, st = 0.f, sq = 0.f;
#pragma unroll
    for (int i = 0; i < ITERS; ++i) {
        f32x4 a = __builtin_nontemporal_load(in4 + i * THREADS);
        f32x4 b = __builtin_nontemporal_load(tg4 + i * THREADS);
#pragma unroll
        for (int k = 0; k < 4; ++k) {
            float d = a[k] - b[k];
            si += a[k];
            st += b[k];
            sq = fmaf(d, d, sq);
        }
    }

    // intra-wave reduce (wave32)
#pragma unroll
    for (int off = 16; off > 0; off >>= 1) {
        si += __shfl_down(si, off, 32);
        st += __shfl_down(st, off, 32);
        sq += __shfl_down(sq, off, 32);
    }

    __shared__ float red[3][THREADS / 32];
    const int wid  = threadIdx.x >> 5;
    const int lane = threadIdx.x & 31;
    if (lane == 0) { red[0][wid] = si; red[1][wid] = st; red[2][wid] = sq; }
    __syncthreads();

    if (threadIdx.x == 0) {
        float a = 0.f, b = 0.f, c = 0.f;
#pragma unroll
        for (int w = 0; w < THREADS / 32; ++w) {
            a += red[0][w]; b += red[1][w]; c += red[2][w];
        }
        const int slot = ch * BLK_PER_CH + bx;   // unique per block -> deterministic
        ws[slot]             = a;
        ws[SLOTS + slot]     = b;
        ws[2 * SLOTS + slot] = c;
    }
}

__global__ void focal_finalize(const float* __restrict__ ws,
                               float* __restrict__ out) {
    const int lane = threadIdx.x;       // 0..31, single wave, EXEC all ones

    float partial = 0.f;
#pragma unroll
    for (int k = 0; k < 3; ++k) {
        const int c = lane + 32 * k;    // covers channels 0..95
        float pre = 0.f, gt = 0.f, sq = 0.f;
#pragma unroll 8
        for (int b = 0; b < BLK_PER_CH; ++b) {
            pre += ws[c * BLK_PER_CH + b];
            gt  += ws[SLOTS + c * BLK_PER_CH + b];
            sq  += ws[2 * SLOTS + c * BLK_PER_CH + b];
        }
        const float mse = sq * (1.0f / (float)HW);
        const float x   = fabsf(pre - gt) / fabsf(gt + 1.0f);
        const float w   = 1.0f + x;      // exp(2*log1p(x)) == (1+x)^2
        partial += mse * w * w;
    }

    // 32-lane sum via V_WMMA_F32_16X16X4_F32:
    // A[m,0] = partial[lane m], A[m,2] = partial[lane m+16], A[m,1]=A[m,3]=0.
    // B = ones  =>  D[m,n] = partial[m] + partial[m+16] for every n.
    v2f a;  a.x = partial; a.y = 0.0f;
    v2f bb; bb.x = 1.0f;   bb.y = 1.0f;
    v8f c = {};
    c = __builtin_amdgcn_wmma_f32_16x16x4_f32(
            /*neg_a=*/false, a, /*neg_b=*/false, bb,
            /*c_mod=*/(short)0, c, /*reuse_a=*/false, /*reuse_b=*/false);

    // lanes 0-15 hold rowsums m=0..7 across the 8 D VGPRs; lanes 16-31 hold m=8..15
    float s = 0.f;
#pragma unroll
    for (int r = 0; r < 8; ++r) s += c[r];
    const float other = __shfl(s, 16, 32);
    if (lane == 0) out[0] = s + other;   // = sum of all 32 partials
}

extern "C" void kernel_launch(void* const* d_in, const int* in_sizes, int n_in,
                              void* d_out, int out_size, void* d_ws, size_t ws_size,
                              hipStream_t stream) {
    const float* inp = (const float*)d_in[0];
    const float* tgt = (const float*)d_in[1];
    float* ws  = (float*)d_ws;   // needs 3*96*32*4 = 36864 bytes
    float* out = (float*)d_out;

    dim3 grid(BLK_PER_CH, NCH);
    focal_partial<<<grid, THREADS, 0, stream>>>(inp, tgt, ws);
    focal_finalize<<<1, 32, 0, stream>>>(ws, out);
}